// DownSampler_21225728377362
// MI455X (gfx1250) — compile-verified
//
#include <hip/hip_runtime.h>
#include <math.h>

// ---------------- problem constants (match reference) ----------------
#define N_PTS   65536
#define N_KEEP  16384
#define D_IN    256
#define D_OUT   256
#define KNN_K   6
#define E1      (N_KEEP * KNN_K)          // 98304 directed knn edges
#define E_TOT   (2 * E1)                  // 196608 after force_undirected

// d_out layout (float elements), concatenated in reference return order
#define OFF_XC    0
#define OFF_POSC  (N_KEEP * D_OUT)        // 4194304
#define OFF_EIDX  (OFF_POSC + N_KEEP * 3) // 4243456
#define OFF_EATTR (OFF_EIDX + 2 * E_TOT)  // 4636672

typedef float v2f __attribute__((ext_vector_type(2)));
typedef float v8f __attribute__((ext_vector_type(8)));

// ---------------------------------------------------------------------
// 1) score[i] = dot(x[i,:], sel_weight)  (softmax is monotonic -> skip)
//    one wave32 per row; lane loads 8 contiguous floats; shfl-xor reduce
// ---------------------------------------------------------------------
__global__ __launch_bounds__(256) void score_init_kernel(
    const float* __restrict__ x, const float* __restrict__ w,
    float* __restrict__ keys, int* __restrict__ idx) {
  const int wave = threadIdx.x >> 5;
  const int lane = threadIdx.x & 31;
  const int row  = blockIdx.x * 8 + wave;        // 8192 blocks * 8 rows
  const float* xr = x + (size_t)row * D_IN + lane * 8;
  const float* wr = w + lane * 8;
  float s = 0.f;
#pragma unroll
  for (int t = 0; t < 8; ++t) s += xr[t] * wr[t];
#pragma unroll
  for (int off = 16; off >= 1; off >>= 1) s += __shfl_xor(s, off, 32);
  if (lane == 0) { keys[row] = s; idx[row] = row; }
}

// ---------------------------------------------------------------------
// 2) bitonic sort pass over 65536 (key,idx), final order:
//    key descending, index ascending on ties (emulates jax.lax.top_k)
// ---------------------------------------------------------------------
__global__ __launch_bounds__(256) void bitonic_pass_kernel(
    float* __restrict__ keys, int* __restrict__ idx,
    unsigned j, unsigned k) {
  const unsigned i = blockIdx.x * blockDim.x + threadIdx.x;
  const unsigned ixj = i ^ j;
  if (ixj <= i) return;
  const float ka = keys[i], kb = keys[ixj];
  const int   ia = idx[i],  ib = idx[ixj];
  const bool a_first = (ka > kb) || (ka == kb && ia < ib); // a belongs before b
  const bool dir = ((i & k) == 0);   // this run sorted in final order?
  const bool sw  = dir ? !a_first : a_first;
  if (sw) { keys[i] = kb; keys[ixj] = ka; idx[i] = ib; idx[ixj] = ia; }
}

// ---------------------------------------------------------------------
// 3) gather pos_c = pos[perm]; emit to d_out and padded float4 scratch
// ---------------------------------------------------------------------
__global__ __launch_bounds__(256) void gather_pos_kernel(
    const float* __restrict__ pos, const int* __restrict__ perm,
    float* __restrict__ out_posc, float4* __restrict__ posc4) {
  const int i = blockIdx.x * blockDim.x + threadIdx.x;   // < 16384
  const int p = perm[i];
  const float px = pos[p * 3 + 0];
  const float py = pos[p * 3 + 1];
  const float pz = pos[p * 3 + 2];
  out_posc[i * 3 + 0] = px;
  out_posc[i * 3 + 1] = py;
  out_posc[i * 3 + 2] = pz;
  posc4[i] = make_float4(px, py, pz, 0.f);
}

// ---------------------------------------------------------------------
// 4) x_c = x[perm] @ lin_w^T + lin_b  via V_WMMA_F32_16X16X4_F32
//    One wave computes a 16x64 output slab: 4 accumulators share one A
//    fragment per K-step (4x less A traffic, 4 wmma per 5 loads).
//    Layout per CDNA5 ISA:
//      A 16x4: lanes 0-15 K={0,1}, lanes 16-31 K={2,3}, M = lane&15
//      B 4x16: mirrored (N = lane&15), B[k,n] = lin_w[n,k] -> float2 loads
//      C/D:    M = r + 8*(lane>=16), N = lane&15
// ---------------------------------------------------------------------
__global__ __launch_bounds__(256) void xc_wmma_kernel(
    const float* __restrict__ x, const float* __restrict__ lin_w,
    const float* __restrict__ lin_b, const int* __restrict__ perm,
    float* __restrict__ out_xc) {
  const int wave = threadIdx.x >> 5;
  const int lane = threadIdx.x & 31;
  const int tile = blockIdx.x * 8 + wave;   // 4096 slabs (1024 M x 4 N)
  const int m0 = (tile >> 2) << 4;          // 16-row band
  const int n0 = (tile & 3) << 6;           // 64-col slab
  const int lrow = lane & 15;
  const int koff = (lane >> 4) << 1;        // 0 or 2

  const float* arow = x + (size_t)perm[m0 + lrow] * D_IN + koff;
  const float* brow = lin_w + (size_t)(n0 + lrow) * D_IN + koff;

  v8f acc0 = {}, acc1 = {}, acc2 = {}, acc3 = {};
#pragma unroll 4
  for (int kk = 0; kk < D_IN; kk += 4) {
    const float2 av = *(const float2*)(arow + kk);
    const float2 b0 = *(const float2*)(brow + kk);
    const float2 b1 = *(const float2*)(brow + 16 * D_IN + kk);
    const float2 b2 = *(const float2*)(brow + 32 * D_IN + kk);
    const float2 b3 = *(const float2*)(brow + 48 * D_IN + kk);
    v2f a;  a[0] = av.x;  a[1] = av.y;
    v2f v0; v0[0] = b0.x; v0[1] = b0.y;
    v2f v1; v1[0] = b1.x; v1[1] = b1.y;
    v2f v2; v2[0] = b2.x; v2[1] = b2.y;
    v2f v3; v3[0] = b3.x; v3[1] = b3.y;
    acc0 = __builtin_amdgcn_wmma_f32_16x16x4_f32(false, a, false, v0,
                                                 (short)0, acc0, false, false);
    acc1 = __builtin_amdgcn_wmma_f32_16x16x4_f32(false, a, false, v1,
                                                 (short)0, acc1, false, false);
    acc2 = __builtin_amdgcn_wmma_f32_16x16x4_f32(false, a, false, v2,
                                                 (short)0, acc2, false, false);
    acc3 = __builtin_amdgcn_wmma_f32_16x16x4_f32(false, a, false, v3,
                                                 (short)0, acc3, false, false);
  }

  const int half8 = (lane >> 4) << 3;
#pragma unroll
  for (int s = 0; s < 4; ++s) {
    const v8f acc = (s == 0) ? acc0 : (s == 1) ? acc1 : (s == 2) ? acc2 : acc3;
    const int col  = n0 + 16 * s + lrow;
    const float bias = lin_b[col];
#pragma unroll
    for (int r = 0; r < 8; ++r) {
      const int row = m0 + r + half8;
      out_xc[(size_t)row * D_OUT + col] = acc[r] + bias;
    }
  }
}

// ---------------------------------------------------------------------
// 5) brute-force KNN (k=6) over 16384 pos_c points.
//    Candidate tiles are double-buffered in LDS via CDNA5 async loads
//    (GLOBAL_LOAD_ASYNC_TO_LDS_B128, ASYNCcnt-tracked): prefetch tile
//    t+1 while scanning tile t, overlap HBM/L2 latency with VALU work.
// ---------------------------------------------------------------------
__device__ __forceinline__ void async_tile_load(const float4* gsrc,
                                                float4* lds_dst) {
  // LDS destination address = low 32 bits of the generic shared pointer
  const unsigned int loff = (unsigned int)(unsigned long long)lds_dst;
  const unsigned long long gaddr = (unsigned long long)gsrc;
  asm volatile("global_load_async_to_lds_b128 %0, %1, off"
               :: "v"(loff), "v"(gaddr) : "memory");
}

__global__ __launch_bounds__(256) void knn_kernel(
    const float4* __restrict__ posc4, int* __restrict__ nbr) {
  __shared__ float4 sh[2][256];
  const int tid = threadIdx.x;
  const int q = blockIdx.x * blockDim.x + tid;           // < 16384
  const float4 pq = posc4[q];

  float bd[KNN_K];
  int   bi[KNN_K];
#pragma unroll
  for (int s = 0; s < KNN_K; ++s) { bd[s] = INFINITY; bi[s] = 0x7fffffff; }

  const int NTILE = N_KEEP / 256;                        // 64
  async_tile_load(posc4 + tid, &sh[0][tid]);             // tile 0 -> buf 0

  for (int t = 0; t < NTILE; ++t) {
    const int buf = t & 1;
    if (t + 1 < NTILE) {
      async_tile_load(posc4 + (t + 1) * 256 + tid, &sh[(t + 1) & 1][tid]);
      asm volatile("s_wait_asynccnt 0x1" ::: "memory");  // tile t done
    } else {
      asm volatile("s_wait_asynccnt 0x0" ::: "memory");  // last tile done
    }
    __syncthreads();                                     // all waves' data in

    const int cbase = t * 256;
    for (int jj = 0; jj < 256; ++jj) {
      const int c = cbase + jj;
      const float4 pc = sh[buf][jj];
      const float dx = pq.x - pc.x;
      const float dy = pq.y - pc.y;
      const float dz = pq.z - pc.z;
      const float d = dx * dx + dy * dy + dz * dz;
      if (c == q) continue;                              // exclude self-loop
      if (d < bd[KNN_K - 1] || (d == bd[KNN_K - 1] && c < bi[KNN_K - 1])) {
        bd[KNN_K - 1] = d; bi[KNN_K - 1] = c;
#pragma unroll
        for (int s = KNN_K - 1; s > 0; --s) {
          const bool sw = (bd[s] < bd[s - 1]) ||
                          (bd[s] == bd[s - 1] && bi[s] < bi[s - 1]);
          if (sw) {
            const float td = bd[s]; bd[s] = bd[s - 1]; bd[s - 1] = td;
            const int   ti = bi[s]; bi[s] = bi[s - 1]; bi[s - 1] = ti;
          }
        }
      }
    }
    __syncthreads();   // reads done before buffer is overwritten next iter
  }
#pragma unroll
  for (int s = 0; s < KNN_K; ++s) nbr[q * KNN_K + s] = bi[s];
}

// ---------------------------------------------------------------------
// 6) edge_index ([2, 2*E1], values as exact floats) + edge_attr [E,4]
//    senders_u = [nbr, receivers]; receivers_u = [receivers, nbr]
// ---------------------------------------------------------------------
__global__ __launch_bounds__(256) void edges_kernel(
    const int* __restrict__ nbr, const float4* __restrict__ posc4,
    float* __restrict__ out_eidx, float* __restrict__ out_eattr) {
  const int eu = blockIdx.x * blockDim.x + threadIdx.x;  // < 196608
  int su, ru;
  if (eu < E1) { su = nbr[eu]; ru = eu / KNN_K; }
  else         { const int e = eu - E1; su = e / KNN_K; ru = nbr[e]; }
  out_eidx[eu]         = (float)su;   // row 0: senders_u
  out_eidx[E_TOT + eu] = (float)ru;   // row 1: receivers_u
  const float4 ps = posc4[su];
  const float4 pr = posc4[ru];
  const float rx = ps.x - pr.x;
  const float ry = ps.y - pr.y;
  const float rz = ps.z - pr.z;
  out_eattr[eu * 4 + 0] = rx;
  out_eattr[eu * 4 + 1] = ry;
  out_eattr[eu * 4 + 2] = rz;
  out_eattr[eu * 4 + 3] = sqrtf(rx * rx + ry * ry + rz * rz);
}

// ---------------------------------------------------------------------
extern "C" void kernel_launch(void* const* d_in, const int* in_sizes, int n_in,
                              void* d_out, int out_size, void* d_ws, size_t ws_size,
                              hipStream_t stream) {
  (void)in_sizes; (void)n_in; (void)out_size; (void)ws_size;
  const float* x     = (const float*)d_in[0];   // [65536, 256]
  const float* pos   = (const float*)d_in[1];   // [65536, 3]
  const float* selw  = (const float*)d_in[2];   // [256]
  const float* lin_w = (const float*)d_in[3];   // [256, 256]
  const float* lin_b = (const float*)d_in[4];   // [256]
  float* out = (float*)d_out;

  // workspace layout (bytes)
  char* ws = (char*)d_ws;
  float*  keys  = (float*)(ws + 0);             // 65536 f  (256 KB)
  int*    idx   = (int*)  (ws + 262144);        // 65536 i  (256 KB)
  float4* posc4 = (float4*)(ws + 524288);       // 16384 f4 (256 KB)
  int*    nbr   = (int*)  (ws + 786432);        // 98304 i  (384 KB)

  // 1) scores + identity indices
  score_init_kernel<<<N_PTS / 8, 256, 0, stream>>>(x, selw, keys, idx);

  // 2) bitonic sort (descending score, ascending index on ties)
  for (unsigned k = 2; k <= (unsigned)N_PTS; k <<= 1)
    for (unsigned j = k >> 1; j >= 1; j >>= 1)
      bitonic_pass_kernel<<<N_PTS / 256, 256, 0, stream>>>(keys, idx, j, k);
  // perm = idx[0 : N_KEEP]

  // 3) pos_c gather
  gather_pos_kernel<<<N_KEEP / 256, 256, 0, stream>>>(
      pos, idx, out + OFF_POSC, posc4);

  // 4) x_c GEMM via v_wmma_f32_16x16x4_f32 (16x64 slab per wave)
  xc_wmma_kernel<<<(N_KEEP / 16) * (D_OUT / 64) / 8, 256, 0, stream>>>(
      x, lin_w, lin_b, idx, out + OFF_XC);

  // 5) KNN (k=6), async double-buffered LDS tiles
  knn_kernel<<<N_KEEP / 256, 256, 0, stream>>>(posc4, nbr);

  // 6) edges + attrs
  edges_kernel<<<E_TOT / 256, 256, 0, stream>>>(
      nbr, posc4, out + OFF_EIDX, out + OFF_EATTR);
}